// FourierBlock_78116865179795
// MI455X (gfx1250) — compile-verified
//
#include <hip/hip_runtime.h>
#include <hip/hip_bf16.h>
#include <math.h>

// FourierBlock: per-(b,c) series of length 4096: FFT -> keep top-64 |X[f]| (f in [0,2049))
// -> zero rest (Hermitian-symmetric) -> inverse FFT -> real part.
// Memory-bound (128 MB total @ 23.3 TB/s ~ 5.5us floor). FFT stages are 16x16 complex
// DFT-matrix x 16x256 data matmuls on V_WMMA_F32_16X16X4_F32 (full f32 precision).
// Inter-stage twiddles are FUSED into the next stage's B-operand loads (each element is
// read exactly once per stage), using hardware v_sin_f32/v_cos_f32 (angles bounded < 2pi).

typedef __attribute__((ext_vector_type(2))) float v2f;
typedef __attribute__((ext_vector_type(8))) float v8f;

#define B_DIM 32
#define L_DIM 4096
#define C_DIM 128
#define K_TOP 64
#define F_DIM 2049           // rfft bins
#define NTHREADS 256
#define NWAVES 8

__device__ __forceinline__ v2f make_v2f(float x, float y) { v2f r; r.x = x; r.y = y; return r; }

__device__ __forceinline__ v8f wmma_f32_16x16x4(v2f a, v2f b, v8f c) {
  // 8 args: (neg_a, A, neg_b, B, c_mod, C, reuse_a, reuse_b). For F32, NEG[0:1] must be 0.
  return __builtin_amdgcn_wmma_f32_16x16x4_f32(false, a, false, b, (short)0, c, false, false);
}

// One 16-wide column tile of: Y[M][j] = sum_m F[M][m] * (Tw(m,j) * X[m][j]) (complex).
// Dr = fA*Xr + fB*Xi ; Di = fC*Xr + fA*Xi.
// Element address in LDS: idx(m, j) = m*sm + (j%16)*slo + (j/16)*shi, j = tile*16 + laneN.
// If TW, the loaded element (row m) is rotated by angle wtw * m * tmul before the matmul.
template <bool TW>
__device__ __forceinline__ void dft_tile(
    const float* __restrict__ inr, const float* __restrict__ ini,
    float* __restrict__ outr, float* __restrict__ outi,
    int smi, int sloi, int shii,
    int smo, int sloo, int shio,
    const v2f* fA, const v2f* fB, const v2f* fC,
    int laneN, int hi, int tile, float wtw, int tmul)
{
  const int base = laneN * sloi + tile * shii;
  v2f br[4], bi[4];
#pragma unroll
  for (int kc = 0; kc < 4; ++kc) {
    // B-matrix 4x16 slab: lanes 0-15 hold rows K={4kc,4kc+1}, lanes 16-31 rows K={4kc+2,4kc+3}
    const int k0 = 4 * kc + 2 * hi;
    const float r0 = inr[base + k0 * smi],       q0 = ini[base + k0 * smi];
    const float r1 = inr[base + (k0 + 1) * smi], q1 = ini[base + (k0 + 1) * smi];
    if (TW) {
      const float a0 = wtw * (float)(k0 * tmul);        // |a| < 2*pi (indices bounded)
      const float a1 = wtw * (float)((k0 + 1) * tmul);
      const float c0 = __cosf(a0), s0 = __sinf(a0);
      const float c1 = __cosf(a1), s1 = __sinf(a1);
      br[kc] = make_v2f(r0 * c0 - q0 * s0, r1 * c1 - q1 * s1);
      bi[kc] = make_v2f(r0 * s0 + q0 * c0, r1 * s1 + q1 * c1);
    } else {
      br[kc] = make_v2f(r0, r1);
      bi[kc] = make_v2f(q0, q1);
    }
  }
  v8f cr = {0.f, 0.f, 0.f, 0.f, 0.f, 0.f, 0.f, 0.f};
  v8f ci = {0.f, 0.f, 0.f, 0.f, 0.f, 0.f, 0.f, 0.f};
#pragma unroll
  for (int kc = 0; kc < 4; ++kc) {
    cr = wmma_f32_16x16x4(fA[kc], br[kc], cr);
    cr = wmma_f32_16x16x4(fB[kc], bi[kc], cr);
    ci = wmma_f32_16x16x4(fC[kc], br[kc], ci);
    ci = wmma_f32_16x16x4(fA[kc], bi[kc], ci);
  }
  // D 16x16: VGPR v holds row M = v (lanes 0-15) / M = v+8 (lanes 16-31), N = lane%16
  const int obase = laneN * sloo + tile * shio + hi * 8 * smo;
#pragma unroll
  for (int v = 0; v < 8; ++v) {
    outr[obase + v * smo] = cr[v];
    outi[obase + v * smo] = ci[v];
  }
}

// Each of the 8 waves handles tiles {wave, wave+8}: fully unrolled, straight-line WMMA
// (EXEC stays all-ones, no exec-masked loop). TWMODE: 0 = none, 1 = T1 (tmul = tile),
// 2 = T2 (tmul = tile + 16*laneN).
template <int TWMODE>
__device__ __forceinline__ void dft_step(
    const float* __restrict__ inr, const float* __restrict__ ini,
    float* __restrict__ outr, float* __restrict__ outi,
    int smi, int sloi, int shii,
    int smo, int sloo, int shio,
    const v2f* fA, const v2f* fB, const v2f* fC,
    int laneN, int hi, int wave, float sgn)
{
  const float wtw = (TWMODE == 1) ? sgn * (6.2831853071795864769f / 256.0f)
                                  : sgn * (6.2831853071795864769f / 4096.0f);
  const int t0 = wave, t1 = wave + NWAVES;
  const int m0 = (TWMODE == 2) ? (t0 + 16 * laneN) : t0;
  const int m1 = (TWMODE == 2) ? (t1 + 16 * laneN) : t1;
  dft_tile<TWMODE != 0>(inr, ini, outr, outi, smi, sloi, shii, smo, sloo, shio,
                        fA, fB, fC, laneN, hi, t0, wtw, m0);
  dft_tile<TWMODE != 0>(inr, ini, outr, outi, smi, sloi, shii, smo, sloo, shio,
                        fA, fB, fC, laneN, hi, t1, wtw, m1);
}

__global__ __launch_bounds__(NTHREADS)
void fourier_topk_fft_kernel(const float* __restrict__ x, float* __restrict__ out)
{
  __shared__ float s_re[L_DIM];
  __shared__ float s_im[L_DIM];
  __shared__ float s_re2[L_DIM];
  __shared__ float s_im2[L_DIM];
  __shared__ float s_amp[F_DIM];
  __shared__ float s_redV[NTHREADS];
  __shared__ int   s_redI[NTHREADS];
  __shared__ int   s_winner;
  __shared__ unsigned char s_mask[F_DIM];

  const int tid  = threadIdx.x;
  const int bc   = blockIdx.x;          // adjacent blocks = adjacent channels -> L2 reuse of strided lines
  const int c    = bc & (C_DIM - 1);
  const int b    = bc >> 7;
  const int lane = tid & 31;
  const int wave = tid >> 5;
  const int laneN = lane & 15;
  const int hi    = lane >> 4;

  const float* xin = x + (size_t)b * L_DIM * C_DIM + c;
  float* xout      = out + (size_t)b * L_DIM * C_DIM + c;

  // Load series (stride C in global; lines reused across channel-adjacent blocks via 192MB L2)
#pragma unroll
  for (int i = 0; i < L_DIM / NTHREADS; ++i) {
    const int t = tid + NTHREADS * i;
    s_re[t] = xin[(size_t)t * C_DIM];
    s_im[t] = 0.0f;
  }
  for (int f = tid; f < F_DIM; f += NTHREADS) s_mask[f] = 0;

  // Per-lane slabs of the 16-point DFT matrix: A 16x4, lane<16 -> M=lane, K={4kc,4kc+1};
  // lane>=16 -> M=lane-16, K={4kc+2,4kc+3}. F[M][K] = exp(-2pi*i*M*K/16).
  v2f fr[4], fi[4], fin[4];
#pragma unroll
  for (int kc = 0; kc < 4; ++kc) {
    const int k0 = 4 * kc + 2 * hi;
    const int p0 = (laneN * k0) & 15;
    const int p1 = (laneN * (k0 + 1)) & 15;
    const float a0 = -0.39269908169872415481f * (float)p0;   // -2*pi/16 * p
    const float a1 = -0.39269908169872415481f * (float)p1;
    fr[kc]  = make_v2f(__cosf(a0), __cosf(a1));
    fi[kc]  = make_v2f(__sinf(a0), __sinf(a1));
    fin[kc] = make_v2f(-fi[kc].x, -fi[kc].y);
  }
  __syncthreads();

  // ---------------- Forward FFT: x -> X (natural order), twiddles fused ----------------
  // Stage 1: DFT over n2 (stride 256), columns j = n0 + 16*n1
  dft_step<0>(s_re, s_im, s_re2, s_im2, 256, 1, 16, 256, 1, 16, fr, fin, fi, laneN, hi, wave, -1.0f);
  __syncthreads();
  // Stage 2: DFT over n1 (stride 16), columns j = n0 + 16*k2; input pre-rotated by W256^{n1*k2}
  dft_step<1>(s_re2, s_im2, s_re, s_im, 16, 1, 256, 16, 1, 256, fr, fin, fi, laneN, hi, wave, -1.0f);
  __syncthreads();
  // Stage 3: DFT over n0 (stride 1), columns j = k1 + 16*k2; input pre-rotated by W4096^{n0*(k2+16*k1)};
  // output scattered to natural order
  dft_step<2>(s_re, s_im, s_re2, s_im2, 1, 16, 256, 256, 16, 1, fr, fin, fi, laneN, hi, wave, -1.0f);
  __syncthreads();

  // ---------------- Top-64 of |X[f]|, f in [0, 2049) ----------------
  // Each thread owns strip f = tid + 256*i and caches its local (max, argmax); per round only
  // the removed element's owner rescans its strip.
  {
    float bv = -1.0f; int bidx = tid;
    for (int f = tid; f < F_DIM; f += NTHREADS) {
      const float rr = s_re2[f], qq = s_im2[f];
      const float v = rr * rr + qq * qq;   // squared magnitude: same ordering
      s_amp[f] = v;
      if (v > bv) { bv = v; bidx = f; }
    }
    s_redV[tid] = bv; s_redI[tid] = bidx;
  }
  __syncthreads();

  for (int r = 0; r < K_TOP; ++r) {
    if (wave == 0) {
      float v = -2.0f; int idx = 0;
#pragma unroll
      for (int j = 0; j < NTHREADS / 32; ++j) {
        const float vv = s_redV[lane + 32 * j];
        const int   ii = s_redI[lane + 32 * j];
        if (vv > v) { v = vv; idx = ii; }
      }
#pragma unroll
      for (int off = 16; off > 0; off >>= 1) {
        const float ov = __shfl_xor(v, off, 32);
        const int   oi = __shfl_xor(idx, off, 32);
        if (ov > v) { v = ov; idx = oi; }
      }
      if (lane == 0) { s_mask[idx] = 1; s_amp[idx] = -1.0f; s_winner = idx; }
    }
    __syncthreads();
    const int widx = s_winner;
    if ((widx & (NTHREADS - 1)) == tid) {   // owner of the removed element rescans its strip
      float nv = -1.0f; int ni = tid;
      for (int f = tid; f < F_DIM; f += NTHREADS) {
        const float v = s_amp[f];
        if (v > nv) { nv = v; ni = f; }
      }
      s_redV[tid] = nv; s_redI[tid] = ni;
    }
    __syncthreads();
  }

  // ---------------- Mask spectrum + enforce Hermitian symmetry ----------------
#pragma unroll
  for (int i = 0; i < L_DIM / NTHREADS; ++i) {
    const int e = tid + NTHREADS * i;
    if (e < F_DIM && !s_mask[e]) { s_re2[e] = 0.0f; s_im2[e] = 0.0f; }
  }
  __syncthreads();
#pragma unroll
  for (int i = 0; i < L_DIM / NTHREADS; ++i) {
    const int e = tid + NTHREADS * i;
    if (e >= F_DIM) {                      // e in [2049, 4096): mirror of f = 4096 - e
      const int m2 = L_DIM - e;
      s_re2[e] = s_re2[m2];
      s_im2[e] = -s_im2[m2];
    }
  }
  __syncthreads();

  // ---------------- Inverse FFT (conjugated F and twiddles; scale at store) ----------------
  dft_step<0>(s_re2, s_im2, s_re, s_im, 256, 1, 16, 256, 1, 16, fr, fi, fin, laneN, hi, wave, 1.0f);
  __syncthreads();
  dft_step<1>(s_re, s_im, s_re2, s_im2, 16, 1, 256, 16, 1, 256, fr, fi, fin, laneN, hi, wave, 1.0f);
  __syncthreads();
  dft_step<2>(s_re2, s_im2, s_re, s_im, 1, 16, 256, 256, 16, 1, fr, fi, fin, laneN, hi, wave, 1.0f);
  __syncthreads();

  const float inv_n = 1.0f / (float)L_DIM;
#pragma unroll
  for (int i = 0; i < L_DIM / NTHREADS; ++i) {
    const int t = tid + NTHREADS * i;
    xout[(size_t)t * C_DIM] = s_re[t] * inv_n;
  }
}

extern "C" void kernel_launch(void* const* d_in, const int* in_sizes, int n_in,
                              void* d_out, int out_size, void* d_ws, size_t ws_size,
                              hipStream_t stream) {
  (void)in_sizes; (void)n_in; (void)out_size; (void)d_ws; (void)ws_size;
  const float* x = (const float*)d_in[0];
  float* out = (float*)d_out;
  dim3 grid(B_DIM * C_DIM);   // one workgroup per (b, c) series
  dim3 block(NTHREADS);
  fourier_topk_fft_kernel<<<grid, block, 0, stream>>>(x, out);
}